// MultiHeadedAttention_86242943304407
// MI455X (gfx1250) — compile-verified
//
#include <hip/hip_runtime.h>
#include <math.h>

// ---------------------------------------------------------------------------
// MultiHeadedAttention with per-row top-k masked softmax, for gfx1250 (MI455X)
// f32 path via V_WMMA_F32_16X16X4_F32 (wave32 matrix pipe, exact f32 math).
// All WMMA fragments are contiguous 8B-aligned LDS pairs -> ds_load_b64.
// ---------------------------------------------------------------------------

typedef __attribute__((ext_vector_type(2))) float v2f;
typedef __attribute__((ext_vector_type(8))) float v8f;

static constexpr int Bc  = 128;  // batch
static constexpr int Dc  = 256;  // channels
static constexpr int Hc  = 4;    // heads
static constexpr int Nc  = 256;  // points (x)
static constexpr int Mc  = 256;  // points (source)
static constexpr int DIMc = 64;  // per-head dim
static constexpr int PAD  = 66;  // even row stride: 8B-aligned pairs, bank-safe

__device__ __forceinline__ v8f wmma4(v2f a, v2f b, v8f c) {
  // D = A(16x4, f32) * B(4x16, f32) + C(16x16, f32)
  return __builtin_amdgcn_wmma_f32_16x16x4_f32(
      /*neg_a=*/false, a, /*neg_b=*/false, b,
      /*c_mod=*/(short)0, c, /*reuse_a=*/false, /*reuse_b=*/false);
}

__device__ __forceinline__ v2f ld2(const float* p) {
  return *(const v2f*)p;  // 8B-aligned contiguous pair -> ds_load_b64
}

// ---------------------------------------------------------------------------
// Y[b] = W (256x256) @ X[b] (256x256) + bias    (1x1 conv over channel dim)
// Tiles: BM=BN=BK=64, 8 waves/WG; each wave -> 16x32 strip (2 accumulators).
// sW kept row-major [e][f] (A reads pairs along f).
// sX stored transposed [n][f]  (B reads pairs along f for fixed n).
// ---------------------------------------------------------------------------
__global__ __launch_bounds__(256) void gemm_bias_256(
    const float* __restrict__ W,     // [256][256] row-major (e,f)
    const float* __restrict__ bias,  // [256]
    const float* __restrict__ X,     // [B][256][256] (f,n)
    float* __restrict__ Y)           // [B][256][256] (e,n)
{
  __shared__ float sW[64][PAD];   // [e][f]
  __shared__ float sX[64][PAD];   // [n][f]  (transposed)

  const int b  = blockIdx.y;
  const int mt = blockIdx.x >> 2;   // 0..3
  const int nt = blockIdx.x & 3;    // 0..3
  const int e0 = mt * 64, n0 = nt * 64;

  const int t    = threadIdx.x;
  const int lane = t & 31, w = t >> 5;
  const int wm = w & 3, wn = w >> 2;        // wave tile: rows 16*wm, cols 32*wn
  const int lrow = lane & 15, lhi = lane >> 4;

  const float* Xb = X + (size_t)b * Dc * Nc;
  v8f acc0 = {};
  v8f acc1 = {};

  for (int f0 = 0; f0 < Dc; f0 += 64) {
#pragma unroll
    for (int i = 0; i < 16; ++i) {            // 256 thr * 16 = 4096 elems
      int idx = t + i * 256;
      int r = idx >> 6, c = idx & 63;         // r: f (or e), c fast (coalesced)
      sW[r][c] = W[(size_t)(e0 + r) * Dc + f0 + c];
      sX[c][r] = Xb[(size_t)(f0 + r) * Nc + n0 + c];   // transpose on store
    }
    __syncthreads();

#pragma unroll
    for (int kk = 0; kk < 64; kk += 4) {
      const int ka = kk + 2 * lhi;            // even -> 8B aligned
      v2f a  = ld2(&sW[16 * wm + lrow][ka]);
      v2f b0 = ld2(&sX[32 * wn + lrow][ka]);
      v2f b1 = ld2(&sX[32 * wn + 16 + lrow][ka]);
      acc0 = wmma4(a, b0, acc0);
      acc1 = wmma4(a, b1, acc1);
    }
    __syncthreads();
  }

  float* Yb = Y + (size_t)b * Dc * Nc;
#pragma unroll
  for (int r = 0; r < 8; ++r) {
    const int e = e0 + 16 * wm + r + 8 * lhi;
    const int n = n0 + 32 * wn + lrow;
    const float bi = bias[e];
    Yb[(size_t)e * Nc + n]      = acc0[r] + bi;
    Yb[(size_t)e * Nc + n + 16] = acc1[r] + bi;
  }
}

// ---------------------------------------------------------------------------
// Attention core per (b, h, 32-row n-block):
//   S = (Q_h^T K_h)/8  -> per-row top-k threshold -> masked softmax -> O = P V^T
// Channel mapping from reshape(b, DIM, H, n): channel c = d*H + h.
// LDS layouts (all fragments contiguous pairs):
//   sQ  [n][d]   A pairs along d           (phase 1)
//   sKV [m][d]   B pairs along d           (phase 1, K transposed on store)
//   sKV [d][m]   B pairs along m           (phase 3, V natural layout)
//   sS  [n][m]   A pairs along m           (phase 3), row stride 258 (even)
// O may alias Q: Q fully staged to LDS before any O write; each WG writes
// exactly the (b,h,n-block) region only it reads.
// ---------------------------------------------------------------------------
__global__ __launch_bounds__(256) void attn_topk(
    const float* __restrict__ Q,   // [B][256][256]
    const float* __restrict__ Kb,  // [B][256][256]
    const float* __restrict__ V,   // [B][256][256]
    const int* __restrict__ kptr,  // top-k (runtime scalar, device mem)
    float* __restrict__ O)         // [B][256][256]
{
  constexpr int RN = 32;                    // n-rows per workgroup
  __shared__ float sQ[RN][PAD];             // [n][d]  (Q transposed)
  __shared__ float sKV[DIMc][PAD];          // phase1: [m][d]; phase3: [d][m]
  __shared__ float sS[RN][Mc + 2];          // scores -> probs, stride 258

  const int nb = blockIdx.x;                // 0..7
  const int h  = blockIdx.y;
  const int b  = blockIdx.z;
  const int n0 = nb * RN;

  const int t    = threadIdx.x;
  const int lane = t & 31, w = t >> 5;
  const int lrow = lane & 15, lhi = lane >> 4;
  const int kTop = *kptr;

  const size_t base = (size_t)b * Dc * Nc;

  // ---- stage Q tile: sQ[n][d] = Q[b][(d*H+h)][n0+n] ----
#pragma unroll
  for (int i = 0; i < (RN * DIMc) / 256; ++i) {   // 8 iters
    int idx = t + i * 256;
    int d = idx >> 5, n = idx & 31;               // n fast (coalesced)
    sQ[n][d] = Q[base + (size_t)(d * Hc + h) * Nc + n0 + n];
  }
  __syncthreads();

  // ---- phase 1: scores. wave tile: tn in {0,1} x tm in {0..3} per 64-col blk
  const int tn = w & 1;
  const int tm = w >> 1;
  for (int m0 = 0; m0 < Mc; m0 += 64) {
#pragma unroll
    for (int i = 0; i < 16; ++i) {
      int idx = t + i * 256;
      int d = idx >> 6, m = idx & 63;             // m fast (coalesced)
      sKV[m][d] = Kb[base + (size_t)(d * Hc + h) * Mc + m0 + m];  // transpose
    }
    __syncthreads();

    v8f acc = {};
#pragma unroll
    for (int kk = 0; kk < DIMc; kk += 4) {
      const int ka = kk + 2 * lhi;
      v2f a  = ld2(&sQ[16 * tn + lrow][ka]);
      v2f bb = ld2(&sKV[16 * tm + lrow][ka]);
      acc = wmma4(a, bb, acc);
    }
#pragma unroll
    for (int r = 0; r < 8; ++r)
      sS[16 * tn + r + 8 * lhi][m0 + 16 * tm + lrow] = acc[r] * 0.125f;
    __syncthreads();
  }

  // ---- phase 2: per-row top-k threshold + masked softmax (4 rows per wave)
  for (int rr = 0; rr < 4; ++rr) {
    const int row = w * 4 + rr;
    float vals[8], work[8];
#pragma unroll
    for (int j = 0; j < 8; ++j) {
      vals[j] = sS[row][lane + 32 * j];
      work[j] = vals[j];
    }
    float thr = 0.0f, rowmax = -__builtin_inff();
    for (int it = 0; it < kTop; ++it) {
      float lm = work[0];
#pragma unroll
      for (int j = 1; j < 8; ++j) lm = fmaxf(lm, work[j]);
      float gm = lm;
#pragma unroll
      for (int off = 16; off > 0; off >>= 1)
        gm = fmaxf(gm, __shfl_xor(gm, off, 32));
      if (it == 0) rowmax = gm;
      thr = gm;
      // remove exactly one instance of the current max (duplicate-aware kth)
      unsigned long long msk = __ballot(lm == gm);
      int firstLane = __ffsll(msk) - 1;
      if (lane == firstLane) {
#pragma unroll
        for (int j = 0; j < 8; ++j) {
          if (work[j] == gm) { work[j] = -__builtin_inff(); break; }
        }
      }
    }
    float e[8];
    float sum = 0.0f;
#pragma unroll
    for (int j = 0; j < 8; ++j) {
      const bool keep = vals[j] >= thr;
      e[j] = keep ? __expf(vals[j] - rowmax) : 0.0f;
      sum += e[j];
    }
#pragma unroll
    for (int off = 16; off > 0; off >>= 1) sum += __shfl_xor(sum, off, 32);
    const float inv = 1.0f / sum;
#pragma unroll
    for (int j = 0; j < 8; ++j) sS[row][lane + 32 * j] = e[j] * inv;
  }
  __syncthreads();

  // ---- phase 3: O^T[n][d] = sum_m P[n][m] * V[d][m]; one 16x16 tile per wave
  const int td = w >> 1;                    // d tile 0..3, reuse tn for n tile
  v8f acc = {};
  for (int m0 = 0; m0 < Mc; m0 += 64) {
#pragma unroll
    for (int i = 0; i < 16; ++i) {          // stage V natural: sKV[d][m]
      int idx = t + i * 256;
      int d = idx >> 6, m = idx & 63;       // m fast: coalesced global AND LDS
      sKV[d][m] = V[base + (size_t)(d * Hc + h) * Mc + m0 + m];
    }
    __syncthreads();
#pragma unroll
    for (int kk = 0; kk < 64; kk += 4) {
      const int ka = kk + 2 * lhi;
      v2f a  = ld2(&sS[16 * tn + lrow][m0 + ka]);
      v2f bb = ld2(&sKV[16 * td + lrow][ka]);
      acc = wmma4(a, bb, acc);
    }
    __syncthreads();
  }

#pragma unroll
  for (int r = 0; r < 8; ++r) {
    const int n = n0 + 16 * tn + r + 8 * lhi;
    const int d = 16 * td + lrow;
    O[base + (size_t)(d * Hc + h) * Nc + n] = acc[r];
  }
}

// ---------------------------------------------------------------------------
// Host-side launcher
// ---------------------------------------------------------------------------
extern "C" void kernel_launch(void* const* d_in, const int* in_sizes, int n_in,
                              void* d_out, int out_size, void* d_ws,
                              size_t ws_size, hipStream_t stream) {
  (void)in_sizes; (void)n_in; (void)out_size; (void)ws_size;

  const float* x   = (const float*)d_in[0];
  const float* src = (const float*)d_in[1];
  const float* Wq  = (const float*)d_in[2];
  const float* bq  = (const float*)d_in[3];
  const float* Wk  = (const float*)d_in[4];
  const float* bk  = (const float*)d_in[5];
  const float* Wv  = (const float*)d_in[6];
  const float* bv  = (const float*)d_in[7];
  const float* Wm  = (const float*)d_in[8];
  const float* bm  = (const float*)d_in[9];
  const int*   kk  = (const int*)d_in[10];
  float* out = (float*)d_out;

  const size_t per = (size_t)Bc * Dc * Nc;   // 8,388,608 floats each
  float* bufQ = (float*)d_ws;                // Q, later re-used as O
  float* bufK = bufQ + per;
  float* bufV = bufK + per;

  dim3 blk(256);
  dim3 gg(16, Bc);        // 4 M-tiles x 4 N-tiles, per batch
  gemm_bias_256<<<gg, blk, 0, stream>>>(Wq, bq, x,   bufQ);
  gemm_bias_256<<<gg, blk, 0, stream>>>(Wk, bk, src, bufK);
  gemm_bias_256<<<gg, blk, 0, stream>>>(Wv, bv, src, bufV);

  dim3 ga(Nc / 32, Hc, Bc);   // (8, 4, 128)
  attn_topk<<<ga, blk, 0, stream>>>(bufQ, bufK, bufV, kk, /*O=*/bufQ);

  gemm_bias_256<<<gg, blk, 0, stream>>>(Wm, bm, bufQ, out);
}